// FuzzyNeuron_91328184582493
// MI455X (gfx1250) — compile-verified
//
#include <hip/hip_runtime.h>

typedef __attribute__((ext_vector_type(2))) float v2f;
typedef __attribute__((ext_vector_type(8))) float v8f;

#define R_RULES 256
#define A_FEAT 64
#define B_ROWS 8192
#define RHO_STRIDE 65   // A + 1 (bias in last column)

// Sort 4 values (5-op min/max network) and convert the trapezoid
// (a,b,c,d) into FMA form:
//   rise = fma(x, rs, ro)  with rs = 1/(b-a),  ro = -a/(b-a)
//   fall = fma(x, fs, fo)  with fs = -1/(d-c), fo =  d/(d-c)
__device__ __forceinline__ float4 trap_params(float4 q) {
    float t1 = fminf(q.x, q.y), t2 = fmaxf(q.x, q.y);
    float t3 = fminf(q.z, q.w), t4 = fmaxf(q.z, q.w);
    float a  = fminf(t1, t3);
    float h1 = fmaxf(t1, t3);
    float l2 = fminf(t2, t4);
    float d  = fmaxf(t2, t4);
    float b  = fminf(h1, l2);
    float c  = fmaxf(h1, l2);
    float rs = 1.0f / (b - a);
    float fs = -1.0f / (d - c);
    return make_float4(rs, -a * rs, fs, -d * fs);
}

// XOR-butterfly lane swap via DS_SWIZZLE_B32 (group-of-32 mode):
// dst_lane = ((lane & 0x1f) | 0) ^ xor_mask.  Immediate-only encoding.
__device__ __forceinline__ float swz_xor1(float v) {
    return __int_as_float(__builtin_amdgcn_ds_swizzle(__float_as_int(v), (1 << 10) | 0x1f));
}
__device__ __forceinline__ float swz_xor2(float v) {
    return __int_as_float(__builtin_amdgcn_ds_swizzle(__float_as_int(v), (2 << 10) | 0x1f));
}
__device__ __forceinline__ float swz_xor4(float v) {
    return __int_as_float(__builtin_amdgcn_ds_swizzle(__float_as_int(v), (4 << 10) | 0x1f));
}
__device__ __forceinline__ float swz_xor8(float v) {
    return __int_as_float(__builtin_amdgcn_ds_swizzle(__float_as_int(v), (8 << 10) | 0x1f));
}

__global__ __launch_bounds__(256)
void fuzzy_prep(const float* __restrict__ abcd, float4* __restrict__ params) {
    int t = blockIdx.x * blockDim.x + threadIdx.x;
    if (t < R_RULES * A_FEAT) {
        params[t] = trap_params(((const float4*)abcd)[t]);
    }
}

// One block = 16-row k-tile, 8 waves. Each wave handles two 16x16 (k x r)
// tiles. Membership products first (with wave-uniform early exit on
// all-zero products); the z tile (V_WMMA_F32_16X16X4_F32) only runs when
// some w in the tile is nonzero -- contribution is exactly 0 otherwise.
template <bool PREP>
__global__ __launch_bounds__(256)
void fuzzy_main(const float* __restrict__ x,
                const float* __restrict__ rho,
                const float* __restrict__ pp,
                float* __restrict__ out) {
    __shared__ float xs[16 * A_FEAT];
    __shared__ float num_s[16];
    __shared__ float den_s[16];

    const int t  = threadIdx.x;
    const int k0 = blockIdx.x * 16;

    // Stage x tile (16 rows x 64 cols, contiguous in global) into LDS.
    for (int i = t; i < 16 * A_FEAT; i += 256) {
        xs[i] = x[k0 * A_FEAT + i];
    }
    if (t < 16) { num_s[t] = 0.0f; den_s[t] = 0.0f; }
    __syncthreads();

    const int lane = t & 31;
    const int wave = t >> 5;
    const int half = lane >> 4;   // which half-wave (K-split for A/B frags)
    const int ln   = lane & 15;   // M for A-frag, N (rule col) for B/C

    // A fragments: 16x4 f32 tiles. Lane ln holds row M=ln; half 0 holds
    // K=4c+{0,1}, half 1 holds K=4c+{2,3}  (ISA 7.12.2 layout).
    v2f afrag[16];
    {
        const float* xr = xs + ln * A_FEAT + 2 * half;
        #pragma unroll
        for (int c = 0; c < 16; ++c) {
            afrag[c][0] = xr[4 * c + 0];
            afrag[c][1] = xr[4 * c + 1];
        }
    }

    // Base row of the C/D layout this half-wave owns (M = v + 8*half).
    const float* xrow = xs + (half * 8) * A_FEAT;

    // Per-row partial sums over this wave's 32 rules (2 r-tiles).
    float nv[8], dv[8];
    #pragma unroll
    for (int v = 0; v < 8; ++v) { nv[v] = 0.0f; dv[v] = 0.0f; }

    #pragma unroll
    for (int it = 0; it < 2; ++it) {
        const int rt   = wave + 8 * it;      // 8 waves x 2 = 16 rule tiles
        const int rule = rt * 16 + ln;       // this lane's rule (N column)
        const float* rrow = rho + rule * RHO_STRIDE;

        // ---- membership products for the 8 rows this lane holds ----
        float w[8];
        #pragma unroll
        for (int v = 0; v < 8; ++v) w[v] = 1.0f;

        const float4* prow = ((const float4*)pp) + rule * A_FEAT;
        __builtin_prefetch(prow, 0, 1);

        for (int ac = 0; ac < A_FEAT; ac += 4) {
            float4 P[4];
            #pragma unroll
            for (int j = 0; j < 4; ++j) {
                float4 raw = prow[ac + j];
                P[j] = PREP ? raw : trap_params(raw);
            }
            #pragma unroll
            for (int v = 0; v < 8; ++v) {
                float4 X = *(const float4*)(xrow + v * A_FEAT + ac); // ds_load_b128
                float wv = w[v];
                {
                    float rise = __builtin_fmaf(X.x, P[0].x, P[0].y);
                    float fall = __builtin_fmaf(X.x, P[0].z, P[0].w);
                    wv *= fmaxf(fminf(fminf(rise, 1.0f), fall), 0.0f);
                }
                {
                    float rise = __builtin_fmaf(X.y, P[1].x, P[1].y);
                    float fall = __builtin_fmaf(X.y, P[1].z, P[1].w);
                    wv *= fmaxf(fminf(fminf(rise, 1.0f), fall), 0.0f);
                }
                {
                    float rise = __builtin_fmaf(X.z, P[2].x, P[2].y);
                    float fall = __builtin_fmaf(X.z, P[2].z, P[2].w);
                    wv *= fmaxf(fminf(fminf(rise, 1.0f), fall), 0.0f);
                }
                {
                    float rise = __builtin_fmaf(X.w, P[3].x, P[3].y);
                    float fall = __builtin_fmaf(X.w, P[3].z, P[3].w);
                    wv *= fmaxf(fminf(fminf(rise, 1.0f), fall), 0.0f);
                }
                w[v] = wv;
            }
            // Early exit: trapezoid factors clamp at exactly 0.0f, so a
            // dead product stays dead. If every (k,r) entry this wave
            // owns is dead, the remaining features cannot change num/den.
            float m01 = fmaxf(w[0], w[1]), m23 = fmaxf(w[2], w[3]);
            float m45 = fmaxf(w[4], w[5]), m67 = fmaxf(w[6], w[7]);
            float wmax = fmaxf(fmaxf(m01, m23), fmaxf(m45, m67));
            if (!__any(wmax > 0.0f)) break;   // wave-uniform s_cbranch
        }

        // ---- tile liveness gate for the z GEMM ----
        float m01 = fmaxf(w[0], w[1]), m23 = fmaxf(w[2], w[3]);
        float m45 = fmaxf(w[4], w[5]), m67 = fmaxf(w[6], w[7]);
        float wmax = fmaxf(fmaxf(m01, m23), fmaxf(m45, m67));
        if (__any(wmax > 0.0f)) {
            // z tile: 16 chained WMMA f32 16x16x4 over the K=64 dim.
            // Branch is wave-uniform, so EXEC is all-1s here (WMMA req).
            v8f acc = {};
            #pragma unroll
            for (int c = 0; c < 16; ++c) {
                v2f bfrag;
                bfrag[0] = rrow[4 * c + 2 * half + 0];
                bfrag[1] = rrow[4 * c + 2 * half + 1];
                acc = __builtin_amdgcn_wmma_f32_16x16x4_f32(
                    /*neg_a=*/false, afrag[c],
                    /*neg_b=*/false, bfrag,
                    /*c_mod=*/(short)0, acc,
                    /*reuse_a=*/false, /*reuse_b=*/false);
            }
            const float bias = rrow[A_FEAT];   // rho[:, -1]
            #pragma unroll
            for (int v = 0; v < 8; ++v) {
                nv[v] = __builtin_fmaf(acc[v] + bias, w[v], nv[v]);
                dv[v] += w[v];
            }
        }
    }

    // ---- XOR butterfly over the 16 lanes sharing each C row ----
    // ds_swizzle masks 1/2/4/8 never cross the 16-lane halves on wave32,
    // so lanes 0-15 sum rows v and lanes 16-31 sum rows v+8.
    #pragma unroll
    for (int v = 0; v < 8; ++v) nv[v] += swz_xor1(nv[v]);
    #pragma unroll
    for (int v = 0; v < 8; ++v) dv[v] += swz_xor1(dv[v]);
    #pragma unroll
    for (int v = 0; v < 8; ++v) nv[v] += swz_xor2(nv[v]);
    #pragma unroll
    for (int v = 0; v < 8; ++v) dv[v] += swz_xor2(dv[v]);
    #pragma unroll
    for (int v = 0; v < 8; ++v) nv[v] += swz_xor4(nv[v]);
    #pragma unroll
    for (int v = 0; v < 8; ++v) dv[v] += swz_xor4(dv[v]);
    #pragma unroll
    for (int v = 0; v < 8; ++v) nv[v] += swz_xor8(nv[v]);
    #pragma unroll
    for (int v = 0; v < 8; ++v) dv[v] += swz_xor8(dv[v]);

    // One lane per half-wave commits the 8-row partials for this wave.
    if (ln == 0) {
        #pragma unroll
        for (int v = 0; v < 8; ++v) {
            atomicAdd(&num_s[v + 8 * half], nv[v]);  // ds_add_f32
            atomicAdd(&den_s[v + 8 * half], dv[v]);
        }
    }

    __syncthreads();
    if (t < 16) {
        out[k0 + t] = num_s[t] / (den_s[t] + 1e-13f);
    }
}

extern "C" void kernel_launch(void* const* d_in, const int* in_sizes, int n_in,
                              void* d_out, int out_size, void* d_ws, size_t ws_size,
                              hipStream_t stream) {
    (void)in_sizes; (void)n_in; (void)out_size;
    const float* x    = (const float*)d_in[0];
    const float* abcd = (const float*)d_in[1];
    const float* rho  = (const float*)d_in[2];
    float* out        = (float*)d_out;

    const size_t params_bytes = (size_t)R_RULES * A_FEAT * 4 * sizeof(float);
    if (ws_size >= params_bytes) {
        fuzzy_prep<<<(R_RULES * A_FEAT + 255) / 256, 256, 0, stream>>>(
            abcd, (float4*)d_ws);
        fuzzy_main<true><<<B_ROWS / 16, 256, 0, stream>>>(
            x, rho, (const float*)d_ws, out);
    } else {
        fuzzy_main<false><<<B_ROWS / 16, 256, 0, stream>>>(
            x, rho, abcd, out);
    }
}